// BandedJointEncoder_61057255080463
// MI455X (gfx1250) — compile-verified
//
#include <hip/hip_runtime.h>

typedef __attribute__((ext_vector_type(2))) float v2f;
typedef __attribute__((ext_vector_type(4))) float v4f;
typedef __attribute__((ext_vector_type(8))) float v8f;

constexpr int TT = 256;    // Z_DIM_TIME_LENGTH
constexpr int EE = 64;     // ENCODING_SIZE
constexpr int BB = 32;     // BATCH
constexpr int CH = 3 * EE; // 192 channels

__device__ __forceinline__ float softplus_f(float x) {
    // stable softplus: max(x,0) + log1p(exp(-|x|))
    return fmaxf(x, 0.0f) + log1pf(__expf(-fabsf(x)));
}

// One 256-thread block (8 wave32) per (b, e).
//   scale_tril[r, c] = a[c] * bv[r] for c <= r, else 0   (rank-1 semiseparable)
//   a[c] = 1/cp[c],  bv[r] = cp[r]/d[r],  cp = prefix-prod of (-s/d)
// Transposed 16x16 WMMA tiles: D'[c_local, r_local] = a[c]*bv[r], so each lane
// owns one output ROW and 8 consecutive columns -> two b128 NT stores per tile.
__global__ __launch_bounds__(256, 2)
void banded_joint_encoder_kernel(const float* __restrict__ stat,
                                 float* __restrict__ out_mu,
                                 float* __restrict__ out_st) {
    const int g = blockIdx.x;      // b*EE + e
    const int b = g >> 6;
    const int e = g & 63;
    const int t = threadIdx.x;     // 0..255

    __shared__ float a_lds[TT];
    __shared__ float bv_lds[TT];
    __shared__ float sc[TT];

    // ---- per-t scalar setup -------------------------------------------------
    const float* srow = stat + ((size_t)b * TT + t) * CH;
    __builtin_nontemporal_store(srow[e], out_mu + (size_t)g * TT + t);

    const float xd = srow[EE + 2 * e];
    const float xs = srow[EE + 2 * e + 1];
    const float dg    = softplus_f(xd) + 1.0f;
    const float sp    = softplus_f(xs);
    const float inv_d = 1.0f / dg;
    const float m     = -sp * inv_d;

    // ---- exclusive prefix product cp[t] = prod_{k<t} m[k] -------------------
    sc[t] = m;
    __syncthreads();
    const float v0 = (t == 0) ? 1.0f : sc[t - 1];
    __syncthreads();
    sc[t] = v0;
    __syncthreads();
    #pragma unroll
    for (int off = 1; off < TT; off <<= 1) {
        const float u = (t >= off) ? sc[t - off] : 1.0f;
        const float w = sc[t];
        __syncthreads();
        sc[t] = u * w;
        __syncthreads();
    }
    const float cp = sc[t];
    a_lds[t]  = 1.0f / cp;
    bv_lds[t] = cp * inv_d;
    __syncthreads();

    // ---- tile sweep: wave w owns tile-rows R = 2w, 2w+1 ---------------------
    const int wave = __builtin_amdgcn_readfirstlane(t >> 5);  // SGPR -> scalar branches
    const int lane = t & 31;
    const int rl   = lane & 15;    // row-in-tile this lane stores
    const int half = lane >> 4;    // column half: 0 -> cols 0..7, 1 -> cols 8..15
    const float inf = __builtin_inff();
    const v4f zero4 = {0.0f, 0.0f, 0.0f, 0.0f};

    float* const outm = out_st + (size_t)g * TT * TT;

    #pragma unroll
    for (int rr = 0; rr < 2; ++rr) {
        const int R     = 2 * wave + rr;
        const int rbase = R * 16;

        // B operand (4x16): row factor bv over N = r_local; only K=0 row live.
        const float btmp = bv_lds[rbase + rl];
        v2f Bop; Bop.x = (lane < 16) ? btmp : 0.0f; Bop.y = 0.0f;

        float* const rowp = outm + (size_t)(rbase + rl) * TT + half * 8;

        // ---- strictly-lower tiles: full outer product, finite mask only ----
        for (int C = 0; C < R; ++C) {
            const float atmp = a_lds[C * 16 + rl];
            v2f Aop; Aop.x = (lane < 16) ? atmp : 0.0f; Aop.y = 0.0f;
            v8f Cacc = {};
            v8f D = __builtin_amdgcn_wmma_f32_16x16x4_f32(
                false, Aop, false, Bop, (short)0, Cacc, false, false);
            v4f lo, hi;
            #pragma unroll
            for (int k = 0; k < 4; ++k) {
                const float vlo = D[k], vhi = D[k + 4];
                lo[k] = (fabsf(vlo) < inf) ? vlo : 0.0f;
                hi[k] = (fabsf(vhi) < inf) ? vhi : 0.0f;
            }
            __builtin_nontemporal_store(lo, (v4f*)(rowp + C * 16));
            __builtin_nontemporal_store(hi, (v4f*)(rowp + C * 16 + 4));
        }

        // ---- diagonal tile C == R: triangle + finite mask -------------------
        {
            const int C = R;
            const float atmp = a_lds[C * 16 + rl];
            v2f Aop; Aop.x = (lane < 16) ? atmp : 0.0f; Aop.y = 0.0f;
            v8f Cacc = {};
            v8f D = __builtin_amdgcn_wmma_f32_16x16x4_f32(
                false, Aop, false, Bop, (short)0, Cacc, false, false);
            const int cs = half * 8;
            v4f lo, hi;
            #pragma unroll
            for (int k = 0; k < 4; ++k) {
                float vlo = D[k], vhi = D[k + 4];
                vlo = ((cs + k)     <= rl && fabsf(vlo) < inf) ? vlo : 0.0f;
                vhi = ((cs + k + 4) <= rl && fabsf(vhi) < inf) ? vhi : 0.0f;
                lo[k] = vlo; hi[k] = vhi;
            }
            __builtin_nontemporal_store(lo, (v4f*)(rowp + C * 16));
            __builtin_nontemporal_store(hi, (v4f*)(rowp + C * 16 + 4));
        }

        // ---- strictly-upper tiles: pure zero fill ---------------------------
        for (int C = R + 1; C < 16; ++C) {
            __builtin_nontemporal_store(zero4, (v4f*)(rowp + C * 16));
            __builtin_nontemporal_store(zero4, (v4f*)(rowp + C * 16 + 4));
        }
    }
}

extern "C" void kernel_launch(void* const* d_in, const int* in_sizes, int n_in,
                              void* d_out, int out_size, void* d_ws, size_t ws_size,
                              hipStream_t stream) {
    const float* stat = (const float*)d_in[0];
    float* out = (float*)d_out;
    float* out_mu = out;                           // (32,64,256)
    float* out_st = out + (size_t)BB * EE * TT;    // (32,64,256,256)
    banded_joint_encoder_kernel<<<BB * EE, 256, 0, stream>>>(stat, out_mu, out_st);
}